// involution_58085137711832
// MI455X (gfx1250) — compile-verified
//
#include <hip/hip_runtime.h>
#include <stdint.h>

typedef float v2f __attribute__((ext_vector_type(2)));
typedef float v8f __attribute__((ext_vector_type(8)));

#define C_    256
#define CR    64
#define G_    8
#define GC_   32
#define KS    7
#define KK    49
#define Wd    128
#define Hd    128
#define HW    16384
#define EPS_  1e-5f

#define TH    16
#define TW    16
#define HALO  3
#define XROWS (TH + 2*HALO)    // 22
#define XCOLS (TW + 2*HALO)    // 22
#define XSP   (XROWS * XCOLS)  // 484
#define WTS   50               // wt stride per pixel (49 taps + pad; gcd(50,64)=2)

__global__ __launch_bounds__(256)
void involution_fused(const float* __restrict__ x,
                      const float* __restrict__ w1,
                      const float* __restrict__ b1,
                      const float* __restrict__ gamma_,
                      const float* __restrict__ beta_,
                      const float* __restrict__ mean_,
                      const float* __restrict__ var_,
                      const float* __restrict__ w2,
                      const float* __restrict__ b2,
                      float* __restrict__ out)
{
    // t: 16 N-tiles x 64 k x 16 n, stored as (k/2)*32 + n*2 + (k&1)  -> 64 KB
    __shared__ __align__(16) float t_lds[16 * 64 * 16];
    // per-pixel involution kernels for current group                  -> 51.2 KB
    __shared__ __align__(16) float wt_lds[256 * WTS];
    // x halo tile for current group, [spatial][channel]               -> 62 KB
    __shared__ __align__(16) float xs[XSP * GC_];

    const int tid  = threadIdx.x;
    const int lane = tid & 31;
    const int wave = tid >> 5;
    const int row  = lane & 15;   // M (A) / N (B,C) index
    const int half = lane >> 4;   // 0: lanes 0-15, 1: lanes 16-31

    const int bIdx = blockIdx.x >> 6;
    const int tIdx = blockIdx.x & 63;
    const int h0 = (tIdx >> 3) * TH;
    const int w0 = (tIdx & 7) * TW;

    // ---------------- Phase 1: t = relu(BN(w1 @ x + b1)) ----------------
    for (int nt = 2 * wave; nt < 2 * wave + 2; ++nt) {
        const float* xB = x + (size_t)bIdx * C_ * HW + (size_t)(h0 + nt) * Wd + w0;
        v8f acc[4] = {};
        for (int k = 0; k < C_; k += 4) {
            v2f bm;
            bm.x = xB[(size_t)(k + 2 * half)     * HW + row];
            bm.y = xB[(size_t)(k + 2 * half + 1) * HW + row];
#pragma unroll
            for (int mt = 0; mt < 4; ++mt) {
                v2f am = *(const v2f*)(w1 + (mt * 16 + row) * C_ + k + 2 * half);
                acc[mt] = __builtin_amdgcn_wmma_f32_16x16x4_f32(
                    false, am, false, bm, (short)0, acc[mt], false, false);
            }
        }
#pragma unroll
        for (int mt = 0; mt < 4; ++mt) {
#pragma unroll
            for (int r = 0; r < 8; ++r) {
                int m = mt * 16 + r + half * 8;
                float v  = acc[mt][r] + b1[m];
                float sc = gamma_[m] * rsqrtf(var_[m] + EPS_);
                v = (v - mean_[m]) * sc + beta_[m];
                v = fmaxf(v, 0.f);
                t_lds[nt * 1024 + (m >> 1) * 32 + row * 2 + (m & 1)] = v;
            }
        }
    }

    const int px = tid & 15;
    const int py = tid >> 4;

    // ---------------- Phase 2: per-group GEMM2 + involution ----------------
    for (int g = 0; g < G_; ++g) {
        __syncthreads();   // previous group's involution done reading xs/wt_lds

        // Cooperative load of 22x22x32 x halo tile, zero-padded borders.
        // In-bounds elements: CDNA5 async DMA straight into LDS (ASYNCcnt);
        // out-of-bounds halo: normal zero ds_store. Each element written once.
        {
            const float* xg = x + ((size_t)bIdx * C_ + g * GC_) * HW;
            for (int idx = tid; idx < XSP * GC_; idx += 256) {
                int c  = idx / XSP;
                int s  = idx - c * XSP;
                int rr = s / XCOLS;
                int cc = s - rr * XCOLS;
                int h  = h0 - HALO + rr;
                int w  = w0 - HALO + cc;
                float* ldst = &xs[s * GC_ + c];
                if ((unsigned)h < (unsigned)Hd && (unsigned)w < (unsigned)Wd) {
                    const float* gsrc = xg + (size_t)c * HW + h * Wd + w;
                    unsigned ldsa = (unsigned)(uintptr_t)ldst;
                    asm volatile("global_load_async_to_lds_b32 %0, %1, off"
                                 :: "v"(ldsa), "v"(gsrc) : "memory");
                } else {
                    *ldst = 0.f;
                }
            }
        }

        // GEMM2 (overlaps with the async DMA above):
        // wt = w2[g*49 .. g*49+48] @ t + b2  (rows >= 49 masked)
        for (int nt = 2 * wave; nt < 2 * wave + 2; ++nt) {
            v8f acc[4] = {};
            for (int k4 = 0; k4 < 16; ++k4) {
                v2f bm = *(const v2f*)&t_lds[nt * 1024 + (2 * k4 + half) * 32 + row * 2];
#pragma unroll
                for (int mt = 0; mt < 4; ++mt) {
                    int rrel = mt * 16 + row;
                    v2f am = {};
                    if (rrel < KK)
                        am = *(const v2f*)(w2 + (size_t)(g * KK + rrel) * CR + 4 * k4 + 2 * half);
                    acc[mt] = __builtin_amdgcn_wmma_f32_16x16x4_f32(
                        false, am, false, bm, (short)0, acc[mt], false, false);
                }
            }
#pragma unroll
            for (int mt = 0; mt < 4; ++mt) {
#pragma unroll
                for (int r = 0; r < 8; ++r) {
                    int rrel = mt * 16 + r + half * 8;
                    if (rrel < KK) {
                        float wv = acc[mt][r] + b2[g * KK + rrel];
                        wt_lds[(nt * 16 + row) * WTS + rrel] = wv;
                    }
                }
            }
        }

        // drain this wave's async LDS loads, then make all tiles visible
        asm volatile("s_wait_asynccnt 0x0" ::: "memory");
        __syncthreads();

        // involution: 1 thread per pixel, 32 channels in float4 accumulators
        float4 a8[8];
#pragma unroll
        for (int c = 0; c < 8; ++c) a8[c] = make_float4(0.f, 0.f, 0.f, 0.f);
        float wsum = 0.f;
        const float* wtp = &wt_lds[(py * 16 + px) * WTS];

        for (int i = 0; i < KS; ++i) {
#pragma unroll
            for (int j = 0; j < KS; ++j) {
                float wv = wtp[i * KS + j];
                wsum += wv;
                const float4* xp = (const float4*)&xs[((py + i) * XCOLS + (px + j)) * GC_];
#pragma unroll
                for (int c = 0; c < 8; ++c) {
                    float4 xv = xp[c];
                    a8[c].x = fmaf(wv, xv.x, a8[c].x);
                    a8[c].y = fmaf(wv, xv.y, a8[c].y);
                    a8[c].z = fmaf(wv, xv.z, a8[c].z);
                    a8[c].w = fmaf(wv, xv.w, a8[c].w);
                }
            }
        }

        float* op = out + ((size_t)bIdx * C_ + g * GC_) * HW
                        + (size_t)(h0 + py) * Wd + (w0 + px);
#pragma unroll
        for (int c = 0; c < 8; ++c) {
            op[(size_t)(4 * c + 0) * HW] = a8[c].x * wsum;
            op[(size_t)(4 * c + 1) * HW] = a8[c].y * wsum;
            op[(size_t)(4 * c + 2) * HW] = a8[c].z * wsum;
            op[(size_t)(4 * c + 3) * HW] = a8[c].w * wsum;
        }
    }
}

extern "C" void kernel_launch(void* const* d_in, const int* in_sizes, int n_in,
                              void* d_out, int out_size, void* d_ws, size_t ws_size,
                              hipStream_t stream) {
    const float* x   = (const float*)d_in[0];
    const float* w1  = (const float*)d_in[1];
    const float* b1  = (const float*)d_in[2];
    const float* gg  = (const float*)d_in[3];
    const float* bb  = (const float*)d_in[4];
    const float* mm  = (const float*)d_in[5];
    const float* vv  = (const float*)d_in[6];
    const float* w2  = (const float*)d_in[7];
    const float* b2  = (const float*)d_in[8];
    float* out = (float*)d_out;

    // 8 batches x (128/16)*(128/16) = 512 blocks, 256 threads (8 wave32)
    involution_fused<<<dim3(8 * 64), dim3(256), 0, stream>>>(
        x, w1, b1, gg, bb, mm, vv, w2, b2, out);
}